// GNNStack_89481348645692
// MI455X (gfx1250) — compile-verified
//
#include <hip/hip_runtime.h>

#define N_NODES 50000
#define N_EDGES 800000

typedef __attribute__((ext_vector_type(2))) float v2f;
typedef __attribute__((ext_vector_type(8))) float v8f;

// ---------------------------------------------------------------------------
// Degree / normalization precompute
// ---------------------------------------------------------------------------
__global__ void k_init_deg(float* __restrict__ deg, int n) {
    int i = blockIdx.x * blockDim.x + threadIdx.x;
    if (i < n) deg[i] = 1.0f;  // self-loop contributes 1
}

__global__ void k_count_deg(const int* __restrict__ dst, float* __restrict__ deg, int e) {
    int i = blockIdx.x * blockDim.x + threadIdx.x;
    if (i < e) unsafeAtomicAdd(&deg[dst[i]], 1.0f);  // -> global_atomic_add_f32
}

__global__ void k_dinv(const float* __restrict__ deg, float* __restrict__ dinv, int n) {
    int i = blockIdx.x * blockDim.x + threadIdx.x;
    if (i < n) dinv[i] = rsqrtf(deg[i]);
}

// ---------------------------------------------------------------------------
// fp32 WMMA GEMM: H[M x NC] = X[M x K] @ W[K x NC]
//   - block = NC/16 wave32s; wave w owns output columns [w*16, w*16+16)
//   - block covers ROWS row-tiles (ROWS*16 rows); X staged once in LDS,
//     B panel preloaded once into VGPRs (K/4 v2f fragments per lane),
//     so the inner loop is ds_load_2addr_b64 (A) + v_wmma only.
//   - 50000 = 625 * (5*16): exact, no bounds checks, EXEC stays all-1s.
// ---------------------------------------------------------------------------
template <int K, int NC, int ROWS>
__global__ __launch_bounds__((NC / 16) * 32) void k_gemm_wmma(const float* __restrict__ X,
                                                              const float* __restrict__ W,
                                                              float* __restrict__ H) {
    constexpr int LDX = K + 4;          // 132: bank = (4*row + k) % 64 -> conflict-free
    constexpr int TPB = (NC / 16) * 32; // threads per block
    __shared__ float Xs[ROWS * 16 * LDX];

    const int wave = threadIdx.x >> 5;
    const int lane = threadIdx.x & 31;
    const int half = lane >> 4;  // K-split half of the wave (A/B layout)
    const int l16  = lane & 15;
    const int n0   = wave * 16;
    const int mBase = blockIdx.x * (ROWS * 16);

    // ---- preload B panel into registers: b[j] = W[4j+2h .. +1][n0+l16] ----
    v2f bfrag[K / 4];
#pragma unroll
    for (int j = 0; j < K / 4; ++j) {
        const int kk = 4 * j + half * 2;
        const float* wp = W + kk * NC + n0 + l16;
        bfrag[j].x = wp[0];
        bfrag[j].y = wp[NC];
    }

    // ---- stage X rows [mBase, mBase + ROWS*16) into LDS (float4 coalesced) ----
    for (int idx = threadIdx.x; idx < ROWS * 16 * (K / 4); idx += TPB) {
        const int r = idx / (K / 4);
        const int c = (idx % (K / 4)) * 4;
        float4 v = *reinterpret_cast<const float4*>(X + (unsigned)(mBase + r) * K + c);
        Xs[r * LDX + c + 0] = v.x;
        Xs[r * LDX + c + 1] = v.y;
        Xs[r * LDX + c + 2] = v.z;
        Xs[r * LDX + c + 3] = v.w;
    }
    __syncthreads();

    // ---- ROWS tiles of 16x16 per wave, K unrolled in steps of 4 ----
#pragma unroll
    for (int t = 0; t < ROWS; ++t) {
        const float* xs = &Xs[(t * 16 + l16) * LDX + half * 2];
        v8f acc = {};
#pragma unroll
        for (int j = 0; j < K / 4; ++j) {
            v2f a;
            a.x = xs[4 * j];       // K = 4j + 2h
            a.y = xs[4 * j + 1];   // K = 4j + 2h + 1
            acc = __builtin_amdgcn_wmma_f32_16x16x4_f32(false, a, false, bfrag[j],
                                                        (short)0, acc, false, false);
        }
        // C/D layout: VGPR v -> row (v + half*8), column n0 + l16
        float* hp = H + (unsigned)(mBase + t * 16 + half * 8) * NC + n0 + l16;
#pragma unroll
        for (int v = 0; v < 8; ++v) hp[(unsigned)v * NC] = acc[v];
    }
}

// ---------------------------------------------------------------------------
// Edge scatter: agg[dst] += h[src] * dinv[src]*dinv[dst]   (F/4 threads per edge)
// ---------------------------------------------------------------------------
template <int F>
__global__ void k_edge_agg(const float* __restrict__ h, const int* __restrict__ src,
                           const int* __restrict__ dst, const float* __restrict__ dinv,
                           float* __restrict__ agg, int e) {
    constexpr unsigned TPE = F / 4;  // power of two -> shifts
    unsigned gid  = blockIdx.x * blockDim.x + threadIdx.x;
    unsigned edge = gid / TPE;
    unsigned f    = (gid % TPE) * 4;
    if (edge >= (unsigned)e) return;
    int s = src[edge];
    int d = dst[edge];
    float nrm = dinv[s] * dinv[d];
    float4 hv = *reinterpret_cast<const float4*>(h + (unsigned)s * F + f);
    float* ap = agg + (unsigned)d * F + f;
    unsafeAtomicAdd(ap + 0, hv.x * nrm);
    unsafeAtomicAdd(ap + 1, hv.y * nrm);
    unsafeAtomicAdd(ap + 2, hv.z * nrm);
    unsafeAtomicAdd(ap + 3, hv.w * nrm);
}

// ---------------------------------------------------------------------------
// Epilogue: out = [relu](agg + h/deg + bias)
// ---------------------------------------------------------------------------
template <bool RELU, int F>
__global__ void k_epilogue(const float* __restrict__ agg, const float* __restrict__ h,
                           const float* __restrict__ deg, const float* __restrict__ bias,
                           float* __restrict__ out, int n) {
    unsigned gid = blockIdx.x * blockDim.x + threadIdx.x;
    if (gid >= (unsigned)n * F) return;
    unsigned i = gid / F;  // power of two -> shift
    unsigned f = gid % F;
    float v = agg[gid] + h[gid] * (1.0f / deg[i]) + bias[f];
    if (RELU) v = fmaxf(v, 0.0f);
    out[gid] = v;
}

// ---------------------------------------------------------------------------
// Launch
// ---------------------------------------------------------------------------
extern "C" void kernel_launch(void* const* d_in, const int* in_sizes, int n_in,
                              void* d_out, int out_size, void* d_ws, size_t ws_size,
                              hipStream_t stream) {
    const float* x  = (const float*)d_in[0];
    const int*   ei = (const int*)d_in[1];
    const float* W1 = (const float*)d_in[2];
    const float* b1 = (const float*)d_in[3];
    const float* W2 = (const float*)d_in[4];
    const float* b2 = (const float*)d_in[5];
    const float* W3 = (const float*)d_in[6];
    const float* b3 = (const float*)d_in[7];

    const int N = N_NODES, E = N_EDGES;
    const int* src = ei;       // edge_index[0]
    const int* dst = ei + E;   // edge_index[1]

    float* ws   = (float*)d_ws;
    float* deg  = ws;  ws += N;
    float* dinv = ws;  ws += N;
    float* bufH = ws;  ws += (size_t)N * 128;  // post-GEMM features
    float* bufA = ws;  ws += (size_t)N * 128;  // aggregation accumulator
    float* bufT = ws;  ws += (size_t)N * 128;  // layer activation

    // ---- degree / normalization (shared by all layers) ----
    k_init_deg<<<(N + 255) / 256, 256, 0, stream>>>(deg, N);
    k_count_deg<<<(E + 255) / 256, 256, 0, stream>>>(dst, deg, E);
    k_dinv<<<(N + 255) / 256, 256, 0, stream>>>(deg, dinv, N);

    constexpr int ROWS = 5;                 // 50000 = 625 * 80, exact
    const unsigned gemmBlocks = N / (16 * ROWS);  // 625

    // ---- layer 1: in=128, out=128, relu ----
    k_gemm_wmma<128, 128, ROWS><<<gemmBlocks, 256, 0, stream>>>(x, W1, bufH);
    hipMemsetAsync(bufA, 0, (size_t)N * 128 * sizeof(float), stream);
    k_edge_agg<128><<<((unsigned)E * 32 + 255) / 256, 256, 0, stream>>>(bufH, src, dst,
                                                                        dinv, bufA, E);
    k_epilogue<true, 128><<<((unsigned)N * 128 + 255) / 256, 256, 0, stream>>>(
        bufA, bufH, deg, b1, bufT, N);

    // ---- layer 2: in=128, out=128, relu ----
    k_gemm_wmma<128, 128, ROWS><<<gemmBlocks, 256, 0, stream>>>(bufT, W2, bufH);
    hipMemsetAsync(bufA, 0, (size_t)N * 128 * sizeof(float), stream);
    k_edge_agg<128><<<((unsigned)E * 32 + 255) / 256, 256, 0, stream>>>(bufH, src, dst,
                                                                        dinv, bufA, E);
    k_epilogue<true, 128><<<((unsigned)N * 128 + 255) / 256, 256, 0, stream>>>(
        bufA, bufH, deg, b2, bufT, N);

    // ---- layer 3: in=128, out=64, no relu ----
    k_gemm_wmma<128, 64, ROWS><<<gemmBlocks, 128, 0, stream>>>(bufT, W3, bufH);
    hipMemsetAsync(bufA, 0, (size_t)N * 64 * sizeof(float), stream);
    k_edge_agg<64><<<((unsigned)E * 16 + 255) / 256, 256, 0, stream>>>(bufH, src, dst,
                                                                       dinv, bufA, E);
    k_epilogue<false, 64><<<((unsigned)N * 64 + 255) / 256, 256, 0, stream>>>(
        bufA, bufH, deg, b3, (float*)d_out, N);
}